// MultiplicativeRecursiveNN_46351287058655
// MI455X (gfx1250) — compile-verified
//
#include <hip/hip_runtime.h>
#include <math.h>

#define E 32
#define DEPTH 11
#define N_NODES 2047
#define LN_EPS 1e-5f
#define TS 33   // LDS tile row stride (padded to avoid bank conflicts)

typedef __attribute__((ext_vector_type(2))) float v2f;
typedef __attribute__((ext_vector_type(8))) float v8f;

// ---------------------------------------------------------------------------
// Async global->LDS copy of one 16-byte chunk (CDNA5 GLOBAL_LOAD_ASYNC_TO_LDS,
// tracked by ASYNCcnt; bypasses VGPRs). lds_off is the LDS byte address
// (low 32 bits of the flat pointer to a __shared__ object).
// ---------------------------------------------------------------------------
__device__ inline void async_copy_b128(const float* __restrict__ gsrc,
                                       void* lds_dst) {
  unsigned lds_off = (unsigned)(unsigned long long)lds_dst;
  asm volatile("global_load_async_to_lds_b128 %0, %1, off"
               :: "v"(lds_off), "v"(gsrc)
               : "memory");
}

__device__ inline void wait_async_zero() {
  asm volatile("s_wait_asynccnt 0x0" ::: "memory");
}

// ---------------------------------------------------------------------------
// One 16x16 fp32 output tile of a 32x32 * 32x32 product, accumulated through
// eight V_WMMA_F32_16X16X4_F32 steps (K = 32 in chunks of 4).
// A, B are 32x32 tiles in LDS with row stride TS.
// Fragment layouts per CDNA5 ISA 7.12.2:
//   A 16x4 : lanes 0-15 -> M, regs {K=0,1} / lanes 16-31 regs {K=2,3}
//   B 4x16 : reg r holds row K; lanes 0-15 K=(0,1), lanes 16-31 K=(2,3)
//   C/D    : VGPR r -> rows r (lanes 0-15) and r+8 (lanes 16-31)
// ---------------------------------------------------------------------------
__device__ inline v8f gemm_tile(const float* __restrict__ A,
                                const float* __restrict__ B,
                                int rowBase, int colBase, v8f acc) {
  const int lane = threadIdx.x & 31;
  const int half = lane >> 4;     // 0: lanes 0-15, 1: lanes 16-31
  const int mn   = lane & 15;     // M for A-frag, N for B-frag
#pragma unroll
  for (int k0 = 0; k0 < E; k0 += 4) {
    v2f a, b;
    a.x = A[(rowBase + mn) * TS + k0 + 2 * half + 0];
    a.y = A[(rowBase + mn) * TS + k0 + 2 * half + 1];
    b.x = B[(k0 + 2 * half + 0) * TS + colBase + mn];
    b.y = B[(k0 + 2 * half + 1) * TS + colBase + mn];
    acc = __builtin_amdgcn_wmma_f32_16x16x4_f32(
        /*neg_a=*/false, a, /*neg_b=*/false, b,
        /*c_mod=*/(short)0, acc, /*reuse_a=*/false, /*reuse_b=*/false);
  }
  return acc;
}

__device__ inline void store_tile_lds(float* __restrict__ S,
                                      int rowBase, int colBase, v8f c) {
  const int lane = threadIdx.x & 31;
  const int half = lane >> 4;
  const int n    = lane & 15;
#pragma unroll
  for (int r = 0; r < 8; ++r)
    S[(rowBase + r + 8 * half) * TS + colBase + n] = c[r];
}

// ---------------------------------------------------------------------------
// Leaf kernel: h[leaf] = ReLU(LN(emb_table[word_ids[leaf]]) * gamma + beta)
// One 256-thread block per leaf node (tree indices 1023..2046).
// Vectorized b128 loads: thread t owns elements 4t..4t+3.
// ---------------------------------------------------------------------------
__global__ void __launch_bounds__(256)
leaf_kernel(const int* __restrict__ word_ids,
            const float* __restrict__ emb_table,
            const float* __restrict__ gamma,
            const float* __restrict__ beta,
            float* __restrict__ hbuf) {
  const int node = ((1 << (DEPTH - 1)) - 1) + blockIdx.x;  // 1023 + b
  const float4* src4 = (const float4*)(emb_table +
                                       (long long)word_ids[node] * (E * E));
  const float4 x = src4[threadIdx.x];
  const float4 g = ((const float4*)gamma)[threadIdx.x];
  const float4 bta = ((const float4*)beta)[threadIdx.x];

  float s = x.x + x.y + x.z + x.w;
  float q = x.x * x.x + x.y * x.y + x.z * x.z + x.w * x.w;
  // wave32 reduce
#pragma unroll
  for (int off = 16; off >= 1; off >>= 1) {
    s += __shfl_xor(s, off, 32);
    q += __shfl_xor(q, off, 32);
  }
  __shared__ float rs[8], rq[8];
  const int wid = threadIdx.x >> 5;
  if ((threadIdx.x & 31) == 0) { rs[wid] = s; rq[wid] = q; }
  __syncthreads();
  float ts = 0.f, tq = 0.f;
#pragma unroll
  for (int w = 0; w < 8; ++w) { ts += rs[w]; tq += rq[w]; }
  const float inv_n = 1.0f / (float)(E * E);
  const float mu  = ts * inv_n;
  const float var = tq * inv_n - mu * mu;
  const float rin = rsqrtf(var + LN_EPS);

  float4 y;
  y.x = fmaxf((x.x - mu) * rin * g.x + bta.x, 0.0f);
  y.y = fmaxf((x.y - mu) * rin * g.y + bta.y, 0.0f);
  y.z = fmaxf((x.z - mu) * rin * g.z + bta.z, 0.0f);
  y.w = fmaxf((x.w - mu) * rin * g.w + bta.w, 0.0f);
  ((float4*)(hbuf + (long long)node * (E * E)))[threadIdx.x] = y;
}

// ---------------------------------------------------------------------------
// Combine kernel (one tree level): per node n at level d,
//   T  = h[left] @ h[right]
//   X1 = T @ cur
//   X2 = X1 @ Ww^T + Wb
//   h[n] = ReLU(LN(X2) * gamma + beta)
// One 128-thread block (4 wave32s) per node; each wave owns one 16x16 tile.
// Tile staging uses GLOBAL_LOAD_ASYNC_TO_LDS_B128 (ASYNCcnt path).
// ---------------------------------------------------------------------------
__global__ void __launch_bounds__(128)
combine_kernel(float* __restrict__ hbuf,          // tree-indexed h storage
               const int* __restrict__ word_ids,
               const float* __restrict__ emb_table,
               const float* __restrict__ Ww,
               const float* __restrict__ Wb,
               const float* __restrict__ gamma,
               const float* __restrict__ beta,
               int level_start,                    // s  = 2^d - 1
               int child_start) {                  // cs = 2^(d+1) - 1
  __shared__ float sL[E * TS];   // left child, later overwritten by T
  __shared__ float sR[E * TS];   // right child, later overwritten by X1
  __shared__ float sC[E * TS];   // cur embedding
  __shared__ float sW[E * TS];   // Ww^T stored so that sW[k*TS+j] = Ww[j][k]
  __shared__ float redS[4], redQ[4];

  const int tid  = threadIdx.x;
  const int node = level_start + blockIdx.x;
  const float* hl = hbuf + (long long)(child_start + 2 * blockIdx.x    ) * (E * E);
  const float* hr = hbuf + (long long)(child_start + 2 * blockIdx.x + 1) * (E * E);
  const float* cu = emb_table + (long long)word_ids[node] * (E * E);

  // Async DMA the three 4KB operand tiles into padded LDS.
  // 256 x 16B chunks per tile; chunk c -> row c>>3, cols (c&7)*4..+3
  // (16B chunks never straddle a 128B row, so padding is safe).
#pragma unroll
  for (int it = 0; it < 2; ++it) {
    int c   = tid + 128 * it;
    int row = c >> 3;
    int col = (c & 7) * 4;
    int l   = row * TS + col;
    async_copy_b128(hl + c * 4, &sL[l]);
    async_copy_b128(hr + c * 4, &sR[l]);
    async_copy_b128(cu + c * 4, &sC[l]);
  }
  // Ww needs a transpose, so it goes through VGPRs (overlaps with the DMAs):
  // sW[k][j] = Ww[j][k]  (B-operand layout for X1 @ Ww^T)
#pragma unroll
  for (int it = 0; it < 8; ++it) {           // 1024 elems / 128 threads
    int e   = tid + 128 * it;
    int j = e >> 5, k = e & 31;
    sW[k * TS + j] = Ww[e];
  }
  wait_async_zero();
  __syncthreads();

  const int wid  = tid >> 5;
  const int ti   = (wid >> 1) * 16;          // tile row base
  const int tj   = (wid & 1) * 16;           // tile col base
  const int lane = tid & 31;
  const int half = lane >> 4;
  const int nn   = lane & 15;

  // T = L @ R
  v8f acc = {};
  acc = gemm_tile(sL, sR, ti, tj, acc);
  __syncthreads();                           // all reads of sL/sR done
  store_tile_lds(sL, ti, tj, acc);           // T -> sL
  __syncthreads();

  // X1 = T @ cur
  acc = (v8f){};
  acc = gemm_tile(sL, sC, ti, tj, acc);
  __syncthreads();
  store_tile_lds(sR, ti, tj, acc);           // X1 -> sR
  __syncthreads();

  // X2 = X1 @ Ww^T + Wb
  acc = (v8f){};
  acc = gemm_tile(sR, sW, ti, tj, acc);
  const float wb = Wb[tj + nn];              // bias broadcast along columns
#pragma unroll
  for (int r = 0; r < 8; ++r) acc[r] += wb;

  // LayerNorm over the whole 32x32 matrix + ReLU
  float s = 0.f, q = 0.f;
#pragma unroll
  for (int r = 0; r < 8; ++r) { s += acc[r]; q += acc[r] * acc[r]; }
#pragma unroll
  for (int off = 16; off >= 1; off >>= 1) {
    s += __shfl_xor(s, off, 32);
    q += __shfl_xor(q, off, 32);
  }
  if (lane == 0) { redS[wid] = s; redQ[wid] = q; }
  __syncthreads();
  float ts = redS[0] + redS[1] + redS[2] + redS[3];
  float tq = redQ[0] + redQ[1] + redQ[2] + redQ[3];
  const float inv_n = 1.0f / (float)(E * E);
  const float mu  = ts * inv_n;
  const float var = tq * inv_n - mu * mu;
  const float rin = rsqrtf(var + LN_EPS);

  float* dst = hbuf + (long long)node * (E * E);
#pragma unroll
  for (int r = 0; r < 8; ++r) {
    int m = ti + r + 8 * half;
    int n = tj + nn;
    int e = m * E + n;
    float y = (acc[r] - mu) * rin * gamma[e] + beta[e];
    dst[e] = fmaxf(y, 0.0f);
  }
}

// ---------------------------------------------------------------------------
// Root kernel: logits = root_flat @ Pw^T + Pb; argmax + NLL loss.
// Single 256-thread block (10x1024 GEMV: latency-bound, scalar is fine).
// ---------------------------------------------------------------------------
__global__ void __launch_bounds__(256)
root_kernel(const float* __restrict__ hbuf,
            const float* __restrict__ Pw,
            const float* __restrict__ Pb,
            const int* __restrict__ label,
            float* __restrict__ out) {
  const float* root = hbuf;                  // tree node 0 at offset 0
  float p[10];
#pragma unroll
  for (int l = 0; l < 10; ++l) p[l] = 0.f;
  for (int i = threadIdx.x; i < E * E; i += 256) {
    float v = root[i];
#pragma unroll
    for (int l = 0; l < 10; ++l) p[l] += v * Pw[l * (E * E) + i];
  }
#pragma unroll
  for (int off = 16; off >= 1; off >>= 1)
#pragma unroll
    for (int l = 0; l < 10; ++l) p[l] += __shfl_xor(p[l], off, 32);

  __shared__ float part[10][8];
  const int wid = threadIdx.x >> 5;
  if ((threadIdx.x & 31) == 0)
#pragma unroll
    for (int l = 0; l < 10; ++l) part[l][wid] = p[l];
  __syncthreads();

  if (threadIdx.x == 0) {
    float logits[10];
#pragma unroll
    for (int l = 0; l < 10; ++l) {
      float t = 0.f;
#pragma unroll
      for (int w = 0; w < 8; ++w) t += part[l][w];
      logits[l] = t + Pb[l];
    }
    int best = 0;
    float bm = logits[0];
#pragma unroll
    for (int l = 1; l < 10; ++l)
      if (logits[l] > bm) { bm = logits[l]; best = l; }
    float se = 0.f;
#pragma unroll
    for (int l = 0; l < 10; ++l) se += expf(logits[l] - bm);
    float lse = bm + logf(se);
    int lab = label[0];
    out[0] = (float)best;                    // prediction
    out[1] = -(logits[lab] - lse);           // loss
  }
}

// ---------------------------------------------------------------------------
extern "C" void kernel_launch(void* const* d_in, const int* in_sizes, int n_in,
                              void* d_out, int out_size, void* d_ws, size_t ws_size,
                              hipStream_t stream) {
  const int*   word_ids  = (const int*)  d_in[0];
  const int*   label     = (const int*)  d_in[1];
  const float* emb_table = (const float*)d_in[2];
  const float* Ww        = (const float*)d_in[3];
  const float* Wb        = (const float*)d_in[4];
  const float* gamma     = (const float*)d_in[5];
  const float* beta      = (const float*)d_in[6];
  const float* Pw        = (const float*)d_in[7];
  const float* Pb        = (const float*)d_in[8];
  float* ws  = (float*)d_ws;   // tree-indexed h storage: 2047 * 1024 floats
  float* out = (float*)d_out;

  // Leaves: tree indices 1023..2046
  leaf_kernel<<<1 << (DEPTH - 1), 256, 0, stream>>>(word_ids, emb_table,
                                                    gamma, beta, ws);
  // Internal levels d = 9 .. 0 (true dependency => one launch per level)
  for (int d = DEPTH - 2; d >= 0; --d) {
    int s  = (1 << d) - 1;
    int cs = (1 << (d + 1)) - 1;
    combine_kernel<<<1 << d, 128, 0, stream>>>(ws, word_ids, emb_table,
                                               Ww, Wb, gamma, beta, s, cs);
  }
  root_kernel<<<1, 256, 0, stream>>>(ws, Pw, Pb, label, out);
}